// HashEmbedder_8211977470208
// MI455X (gfx1250) — compile-verified
//
#include <hip/hip_runtime.h>
#include <math.h>

// HashEmbedder forward (Instant-NGP style multi-resolution hash gather).
// Memory-bound: ~264 MB of HBM traffic, ~0.25 G-ops => bandwidth roofline
// (~11 us at 23.3 TB/s). No matmul structure -> no WMMA; optimize the
// data-movement path instead:
//  - b96 coord loads, b64 table gathers (48 MB table lives in the 192 MB L2)
//  - LDS transpose so output stores are 512B-contiguous per wave32 instruction
//  - non-temporal b128 output stores so the streaming 192 MB output does not
//    evict the L2-resident embedding table.

#define NUM_LEVELS 12
#define LOG2_T 19
#define TMASK ((1u << LOG2_T) - 1u)
#define BLOCK 256            // 8 wave32s per block

typedef float v2f __attribute__((ext_vector_type(2)));
typedef float v4f __attribute__((ext_vector_type(4)));

struct ResArr { float r[NUM_LEVELS]; };

__global__ __launch_bounds__(BLOCK) void hashgrid_fwd_kernel(
    const float* __restrict__ x,      // [N,3]
    const float* __restrict__ emb,    // [L, 2^19, 2]
    float* __restrict__ out,          // [N, 24]
    int n, ResArr res)
{
    __shared__ float lds[BLOCK * 2 * NUM_LEVELS];   // 24 KB staging

    const int tid        = threadIdx.x;
    const int blockStart = blockIdx.x * BLOCK;
    const int pt         = blockStart + tid;

    if (pt < n) {
        // 12-byte coordinate fetch (lowers to a single global_load_b96)
        const float* xp = x + 3ull * (unsigned)pt;
        const float cx = xp[0];
        const float cy = xp[1];
        const float cz = xp[2];

        float* w = lds + tid * (2 * NUM_LEVELS);
#pragma unroll
        for (int l = 0; l < NUM_LEVELS; ++l) {
            const float r = res.r[l];
            // coords are in [0, 512): float->uint truncation == floor here,
            // and uint32 wraparound matches the reference's mod-2^32 hash.
            const unsigned x0 = (unsigned)(cx * r);
            const unsigned y0 = (unsigned)(cy * r);
            const unsigned z0 = (unsigned)(cz * r);
            const unsigned h  =
                (x0 ^ (y0 * 2654435761u) ^ (z0 * 805459861u)) & TMASK;
            // 8B gather from the per-level table (L2-resident, default RT hint)
            const v2f e = *(const v2f*)(
                emb + (((size_t)l << LOG2_T) + h) * 2u);
            *(v2f*)(w + 2 * l) = e;   // ds_store_b64
        }
    }
    __syncthreads();

    // Coalesced write-back: each store instruction covers 32 consecutive
    // 16B vectors = 512 contiguous bytes. NT hint keeps the streaming output
    // from thrashing the L2 copy of the embedding table.
    const int vb   = min(BLOCK, n - blockStart);   // valid points in block
    const int nvec = vb * (2 * NUM_LEVELS / 4);    // 16B vector elements
    const v4f* lv = (const v4f*)lds;
    v4f* ov = (v4f*)(out + (size_t)blockStart * (2 * NUM_LEVELS));
#pragma unroll
    for (int k = 0; k < (2 * NUM_LEVELS) / 4; ++k) {
        const int idx = k * BLOCK + tid;
        if (idx < nvec) {
            const v4f v = lv[idx];                       // ds_load_b128
            __builtin_nontemporal_store(v, ov + idx);    // global_store_b128 nt
        }
    }
}

extern "C" void kernel_launch(void* const* d_in, const int* in_sizes, int n_in,
                              void* d_out, int out_size, void* d_ws, size_t ws_size,
                              hipStream_t stream) {
    const float* x   = (const float*)d_in[0];   // [N,3] float32
    const float* emb = (const float*)d_in[1];   // [12, 2^19, 2] float32
    float* out       = (float*)d_out;           // [N, 24] float32

    const int n = in_sizes[0] / 3;

    // Replicate numpy's float64 resolution computation exactly:
    // growth = exp((log(512)-log(16))/11); res_i = int(16 * growth**i)
    ResArr res;
    const double growth = exp((log(512.0) - log(16.0)) / 11.0);
    for (int i = 0; i < NUM_LEVELS; ++i) {
        const double v = 16.0 * pow(growth, (double)i);
        res.r[i] = (float)(long long)v;   // Python int() truncates toward zero
    }

    const int grid = (n + BLOCK - 1) / BLOCK;
    hashgrid_fwd_kernel<<<grid, BLOCK, 0, stream>>>(x, emb, out, n, res);
}